// MoERouter_5677946765396
// MI455X (gfx1250) — compile-verified
//
#include <hip/hip_runtime.h>
#include <math.h>

// ---------------- problem constants ----------------
#define BATCH   4
#define TOKENS  4096
#define NTOK    (BATCH * TOKENS)   // 16384
#define D_DIM   2048
#define E_DIM   64
#define TOPK    2

// ---------------- tiling ----------------
#define ROWS    128                 // tokens per block (8 waves x 16)
#define BK      64                  // K-chunk staged in LDS
#define NCHUNK  (D_DIM / BK)        // 32
#define LS      68                  // LDS row stride in dwords (64 data + 4 pad, kills bank conflicts)

typedef __attribute__((ext_vector_type(2))) float v2f;
typedef __attribute__((ext_vector_type(4))) float v4f;
typedef __attribute__((ext_vector_type(8))) float v8f;
typedef __attribute__((ext_vector_type(4))) int   v4i;

#if defined(__gfx1250__) && __has_builtin(__builtin_amdgcn_global_load_async_to_lds_b128)
#define USE_ASYNC 1
#else
#define USE_ASYNC 0
#endif

__device__ __forceinline__ void async_cp16(const float* g, float* l) {
#if USE_ASYNC
    __builtin_amdgcn_global_load_async_to_lds_b128(
        (__attribute__((address_space(1))) v4i*)g,
        (__attribute__((address_space(3))) v4i*)l,
        0, 0);
#else
    *(v4f*)l = *(const v4f*)g;
#endif
}

__device__ __forceinline__ void wait_async() {
#if USE_ASYNC
#if __has_builtin(__builtin_amdgcn_s_wait_asynccnt)
    __builtin_amdgcn_s_wait_asynccnt(0);
#else
    asm volatile("s_wait_asynccnt 0" ::: "memory");
#endif
#endif
}

__global__ __launch_bounds__(256, 1)
void MoERouter_5677946765396_kernel(const float* __restrict__ x,
                                    const float* __restrict__ gw,
                                    float* __restrict__ wout,
                                    int* __restrict__ iout) {
    // double-buffered staging tiles
    __shared__ float xs[2][ROWS * LS];   // 2 * 128 * 68 * 4B = 69632 B
    __shared__ float gs[2][E_DIM * LS];  // 2 *  64 * 68 * 4B = 34816 B

    const int tid  = threadIdx.x;
    const int lane = tid & 31;
    const int wv   = tid >> 5;               // 8 waves
    const int R0   = blockIdx.x * ROWS;      // first token of this block

    // WMMA 16x16x4 f32 A-fragment addressing:
    //  lanes 0-15 hold rows M=lane with K={0,1}; lanes 16-31 hold same rows with K={2,3}
    const int mrow  = wv * 16 + (lane & 15);
    const int khalf = (lane >> 4) << 1;      // 0 or 2

    v8f acc0 = {}, acc1 = {}, acc2 = {}, acc3 = {};

    auto stage = [&](int buf, int c) {
        const int k0 = c * BK;
        // x tile: 128 rows x 64 f32 = 2048 x 16B chunks -> 8 per thread
        #pragma unroll
        for (int i = 0; i < 8; ++i) {
            int cc  = tid + i * 256;
            int row = cc >> 4;               // 16 chunks per row
            int seg = cc & 15;
            async_cp16(x + (size_t)(R0 + row) * D_DIM + k0 + seg * 4,
                       &xs[buf][row * LS + seg * 4]);
        }
        // gate tile: 64 rows x 64 f32 = 1024 chunks -> 4 per thread
        #pragma unroll
        for (int i = 0; i < 4; ++i) {
            int cc  = tid + i * 256;
            int row = cc >> 4;
            int seg = cc & 15;
            async_cp16(gw + (size_t)row * D_DIM + k0 + seg * 4,
                       &gs[buf][row * LS + seg * 4]);
        }
    };

    stage(0, 0);
    wait_async();
    __syncthreads();

    for (int c = 0; c < NCHUNK; ++c) {
        const int buf = c & 1;
        if (c + 1 < NCHUNK) stage(buf ^ 1, c + 1);   // prefetch overlaps WMMA below

        const float* xrow = &xs[buf][mrow * LS + khalf];
        const float* g0   = &gs[buf][(lane & 15) * LS + khalf];
        const float* g1   = g0 + 16 * LS;
        const float* g2   = g0 + 32 * LS;
        const float* g3   = g0 + 48 * LS;

        #pragma unroll
        for (int kk = 0; kk < BK; kk += 4) {
            v2f a  = *(const v2f*)(xrow + kk);
            v2f b0 = *(const v2f*)(g0 + kk);
            v2f b1 = *(const v2f*)(g1 + kk);
            v2f b2 = *(const v2f*)(g2 + kk);
            v2f b3 = *(const v2f*)(g3 + kk);
            acc0 = __builtin_amdgcn_wmma_f32_16x16x4_f32(false, a, false, b0, (short)0, acc0, false, false);
            acc1 = __builtin_amdgcn_wmma_f32_16x16x4_f32(false, a, false, b1, (short)0, acc1, false, false);
            acc2 = __builtin_amdgcn_wmma_f32_16x16x4_f32(false, a, false, b2, (short)0, acc2, false, false);
            acc3 = __builtin_amdgcn_wmma_f32_16x16x4_f32(false, a, false, b3, (short)0, acc3, false, false);
        }

        wait_async();
        __syncthreads();
    }

    // ---- scatter logits to LDS (reuse xs buffer 0: 128 rows x 64 cols, stride LS) ----
    // C/D layout: VGPR j holds M=j (lanes 0-15) and M=j+8 (lanes 16-31), N=lane&15
    float* lg = &xs[0][0];
    {
        const int col  = lane & 15;
        const int roff = wv * 16 + ((lane >> 4) << 3);
        #pragma unroll
        for (int j = 0; j < 8; ++j) {
            const int row = roff + j;
            lg[row * LS + col +  0] = acc0[j];
            lg[row * LS + col + 16] = acc1[j];
            lg[row * LS + col + 32] = acc2[j];
            lg[row * LS + col + 48] = acc3[j];
        }
    }
    __syncthreads();

    // ---- top-2 + softmax, one thread per token ----
    if (tid < ROWS) {
        const float* r = &lg[tid * LS];
        float s0 = -INFINITY, s1 = -INFINITY;
        int   i0 = 0,         i1 = 0;
        #pragma unroll 8
        for (int e = 0; e < E_DIM; ++e) {
            float v = r[e];
            if (v > s0)      { s1 = s0; i1 = i0; s0 = v; i0 = e; }
            else if (v > s1) { s1 = v;  i1 = e; }
        }
        const float w0 = 1.0f / (1.0f + __expf(s1 - s0));  // softmax of [s0, s1]
        const int tok = R0 + tid;
        wout[2 * tok]     = w0;
        wout[2 * tok + 1] = 1.0f - w0;
        iout[2 * tok]     = i0;
        iout[2 * tok + 1] = i1;
    }
}

extern "C" void kernel_launch(void* const* d_in, const int* in_sizes, int n_in,
                              void* d_out, int out_size, void* d_ws, size_t ws_size,
                              hipStream_t stream) {
    const float* x  = (const float*)d_in[0];   // [B, T, D] f32
    const float* gw = (const float*)d_in[1];   // [E, D]    f32
    float* wout = (float*)d_out;               // weights [B*T*K] f32
    int*   iout = (int*)(wout + NTOK * TOPK);  // indices [B*T*K] i32, concatenated after
    dim3 grid(NTOK / ROWS);                    // 128 blocks
    dim3 block(256);                           // 8 wave32 waves
    MoERouter_5677946765396_kernel<<<grid, block, 0, stream>>>(x, gw, wout, iout);
}